// SSIMLoss_27075473834383
// MI455X (gfx1250) — compile-verified
//
#include <hip/hip_runtime.h>

// ---------------------------------------------------------------------------
// SSIM loss, MI455X (gfx1250): separable 11x11 Gaussian conv expressed as
// banded-matrix WMMA (V_WMMA_F32_16X16X4_F32), LDS transpose between passes,
// deterministic two-phase mean reduction.
// Round 3: wave-uniform interior/boundary specialization of the vertical
// pass. Interior tiles (82%) use one base address + IOFFSET-immediate loads
// (no compares/clamps/cndmasks); only edge tiles pay the masking cost.
// ---------------------------------------------------------------------------

typedef __attribute__((ext_vector_type(2))) float v2f;
typedef __attribute__((ext_vector_type(8))) float v8f;

#define IMG_HW 512
#define SSIM_C1 1.0e-4f      // (0.01*1.0)^2
#define SSIM_C2 9.0e-4f      // (0.03*1.0)^2
#define N_ELEMS 12582912.0f  // 16*3*512*512

__device__ __forceinline__ int clampi(int v, int lo, int hi) {
  return v < lo ? lo : (v > hi ? hi : v);
}

__device__ __forceinline__ v8f wmma_f32(v2f a, v2f b, v8f c) {
  // D = A(16x4,f32) * B(4x16,f32) + C(16x16,f32); 8-arg pattern:
  // (neg_a, A, neg_b, B, c_mod, C, reuse_a, reuse_b)
  return __builtin_amdgcn_wmma_f32_16x16x4_f32(
      false, a, false, b, (short)0, c, false, false);
}

// Vertical pass for one 16-col tile: V(16x16) = Band(16x28) x In(28x16) for
// all five derived planes; result stored to LDS in D layout.
// INTERIOR=true: all addresses in-bounds -> single base pointer, constant
// byte offsets folded into the load's IOFFSET field.
template <bool INTERIOR>
__device__ __forceinline__ void vertical_tile(
    const float* __restrict__ p1, const float* __restrict__ p2,
    int y0, int x0, int ct, int nn, int hh, const v2f* A,
    float* __restrict__ vb /* [5][16][48] */) {
  v8f am1 = {0.f,0.f,0.f,0.f,0.f,0.f,0.f,0.f};
  v8f am2 = am1, a11 = am1, a22 = am1, a12 = am1;

  // Per-lane tile-local origin: row = 2*hh (within K-chunk), col = lane nn.
  const int gx  = x0 - 5 + 16 * ct + nn;
  const bool cx = (unsigned)gx < IMG_HW;
  const int gxc = clampi(gx, 0, IMG_HW - 1);
  const float* __restrict__ q1 =
      p1 + (y0 - 5 + 2 * hh) * IMG_HW + gx;   // only deref'd when INTERIOR
  const float* __restrict__ q2 =
      p2 + (y0 - 5 + 2 * hh) * IMG_HW + gx;

#pragma unroll
  for (int kc = 0; kc < 7; ++kc) {
    // B layout: vgpr0 = row k, vgpr1 = row k+1, with k = 4kc + 2*hh.
    float i1a, i1b, i2a, i2b;
    if (INTERIOR) {
      const int o = kc * 4 * IMG_HW;           // compile-time constant
      i1a = q1[o];
      i1b = q1[o + IMG_HW];
      i2a = q2[o];
      i2b = q2[o + IMG_HW];
    } else {
      const int kA  = 4 * kc + 2 * hh;
      const int gyA = y0 - 5 + kA;
      const int gyB = gyA + 1;
      const bool vA = cx && ((unsigned)gyA < IMG_HW);
      const bool vB = cx && ((unsigned)gyB < IMG_HW);
      const int offA = clampi(gyA, 0, IMG_HW - 1) * IMG_HW + gxc;
      const int offB = clampi(gyB, 0, IMG_HW - 1) * IMG_HW + gxc;
      i1a = p1[offA];
      i1b = p1[offB];
      i2a = p2[offA];
      i2b = p2[offB];
      i1a = vA ? i1a : 0.0f;
      i1b = vB ? i1b : 0.0f;
      i2a = vA ? i2a : 0.0f;
      i2b = vB ? i2b : 0.0f;
    }
    const v2f b1  = { i1a,       i1b       };
    const v2f b2  = { i2a,       i2b       };
    const v2f b11 = { i1a * i1a, i1b * i1b };
    const v2f b22 = { i2a * i2a, i2b * i2b };
    const v2f b12 = { i1a * i2a, i1b * i2b };
    am1 = wmma_f32(A[kc], b1,  am1);
    am2 = wmma_f32(A[kc], b2,  am2);
    a11 = wmma_f32(A[kc], b11, a11);
    a22 = wmma_f32(A[kc], b22, a22);
    a12 = wmma_f32(A[kc], b12, a12);
  }

  // D layout -> LDS: vgpr r holds rows r (lanes 0-15) / r+8 (lanes 16-31).
#pragma unroll
  for (int r = 0; r < 8; ++r) {
    const int idx = (r + 8 * hh) * 48 + 16 * ct + nn;
    vb[0 * 768 + idx] = am1[r];
    vb[1 * 768 + idx] = am2[r];
    vb[2 * 768 + idx] = a11[r];
    vb[3 * 768 + idx] = a22[r];
    vb[4 * 768 + idx] = a12[r];
  }
}

// Phase 1: each wave computes a 16(row) x 32(col) SSIM-map tile of one plane
// and folds it into one float partial per workgroup.
__global__ __launch_bounds__(128) void ssim_tiles_kernel(
    const float* __restrict__ img1, const float* __restrict__ img2,
    const float* __restrict__ window, float* __restrict__ partials) {
  // Per-wave vertical-pass scratch: 5 planes x 16 rows x 48 cols.
  __shared__ float Vb[4][5][16][48];   // 61440 bytes
  __shared__ float wgsum[4];

  const int tid  = threadIdx.x;
  const int wv   = tid >> 5;      // wave in block (0..3)
  const int lane = tid & 31;
  const int nn   = lane & 15;     // M (A) / N (B,D) index
  const int hh   = lane >> 4;     // K half select

  // Tile decode: 48 planes x 32 row-tiles x 16 col-tiles = 24576 tiles.
  const int tile  = blockIdx.x * 4 + wv;
  const int plane = tile >> 9;          // /(32*16)
  const int rem   = tile & 511;
  const int y0    = (rem >> 4) * 16;    // output row base
  const int x0    = (rem & 15) * 32;    // output col base
  const size_t pbase = (size_t)plane * (IMG_HW * IMG_HW);
  const float* __restrict__ p1 = img1 + pbase;
  const float* __restrict__ p2 = img2 + pbase;
  float* __restrict__ vb = &Vb[wv][0][0][0];

  // Recover 1D gaussian from the 2D window: g[j] = W[5][j]/sqrt(W[5][5]).
  // Band matrix A[m][k] = g[k-m] for k-m in [0,10], else 0; K padded to 28
  // (rows 26,27 always weight 0). ISA A-layout: lane m holds row m, lanes
  // 16-31 hold the upper K pair of each 4-chunk.
  const float invs = 1.0f / sqrtf(window[60]);
  v2f A[7];
#pragma unroll
  for (int kc = 0; kc < 7; ++kc) {
    int k0 = 4 * kc + 2 * hh;
    int d0 = k0 - nn;
    int d1 = d0 + 1;
    A[kc].x = (d0 >= 0 && d0 <= 10) ? window[55 + d0] * invs : 0.0f;
    A[kc].y = (d1 >= 0 && d1 <= 10) ? window[55 + d1] * invs : 0.0f;
  }

  // ---- Vertical pass: rows touched = [y0-5, y0+22], cols = [x0-5, x0+42].
  // Interior test is wave-uniform (no barriers inside the branch).
  const bool interior = (y0 >= 5) && (y0 + 22 < IMG_HW) &&
                        (x0 >= 5) && (x0 + 42 < IMG_HW);
  if (interior) {
    for (int ct = 0; ct < 3; ++ct)
      vertical_tile<true>(p1, p2, y0, x0, ct, nn, hh, A, vb);
  } else {
    for (int ct = 0; ct < 3; ++ct)
      vertical_tile<false>(p1, p2, y0, x0, ct, nn, hh, A, vb);
  }
  __syncthreads();

  // ---- Horizontal pass: O_g^T(16x16) = Band(16x28) x V^T slice(28x16),
  // B[k][n] = V[n][16g + k] -> consecutive cols per lane -> ds_load_b64.
  float lsum = 0.0f;
  for (int gI = 0; gI < 2; ++gI) {
    v8f am1 = {0.f,0.f,0.f,0.f,0.f,0.f,0.f,0.f};
    v8f am2 = am1, a11 = am1, a22 = am1, a12 = am1;
#pragma unroll
    for (int kc = 0; kc < 7; ++kc) {
      const int col = 16 * gI + 4 * kc + 2 * hh;   // even -> 8B aligned
      const int row = nn * 48 + col;
      const v2f b1  = *(const v2f*)&vb[0 * 768 + row];
      const v2f b2  = *(const v2f*)&vb[1 * 768 + row];
      const v2f b11 = *(const v2f*)&vb[2 * 768 + row];
      const v2f b22 = *(const v2f*)&vb[3 * 768 + row];
      const v2f b12 = *(const v2f*)&vb[4 * 768 + row];
      am1 = wmma_f32(A[kc], b1,  am1);
      am2 = wmma_f32(A[kc], b2,  am2);
      a11 = wmma_f32(A[kc], b11, a11);
      a22 = wmma_f32(A[kc], b22, a22);
      a12 = wmma_f32(A[kc], b12, a12);
    }
    // SSIM elementwise: all 5 planes share the (transposed) D layout, so
    // per-register elements correspond to the same output pixel.
#pragma unroll
    for (int r = 0; r < 8; ++r) {
      const float mu1 = am1[r], mu2 = am2[r];
      const float x11 = a11[r], x22 = a22[r], x12 = a12[r];
      const float mu1s = mu1 * mu1, mu2s = mu2 * mu2, mu12 = mu1 * mu2;
      const float s1 = x11 - mu1s, s2 = x22 - mu2s, s12 = x12 - mu12;
      const float num = (2.0f * mu12 + SSIM_C1) * (2.0f * s12 + SSIM_C2);
      const float den = (mu1s + mu2s + SSIM_C1) * (s1 + s2 + SSIM_C2);
      lsum += num / den;
    }
  }

  // Wave32 reduction, then fixed-order block combine (deterministic).
#pragma unroll
  for (int off = 16; off > 0; off >>= 1)
    lsum += __shfl_xor(lsum, off, 32);
  if (lane == 0) wgsum[wv] = lsum;
  __syncthreads();
  if (tid == 0)
    partials[blockIdx.x] = (wgsum[0] + wgsum[1]) + (wgsum[2] + wgsum[3]);
}

// Phase 2: single block, fixed-order tree -> deterministic scalar.
__global__ __launch_bounds__(256) void ssim_reduce_kernel(
    const float* __restrict__ partials, int nparts, float* __restrict__ out) {
  __shared__ float sm[256];
  float t = 0.0f;
  for (int i = threadIdx.x; i < nparts; i += 256) t += partials[i];
  sm[threadIdx.x] = t;
  __syncthreads();
  for (int off = 128; off > 0; off >>= 1) {
    if ((int)threadIdx.x < off) sm[threadIdx.x] += sm[threadIdx.x + off];
    __syncthreads();
  }
  if (threadIdx.x == 0) out[0] = 1.0f - sm[0] * (1.0f / N_ELEMS);
}

extern "C" void kernel_launch(void* const* d_in, const int* in_sizes, int n_in,
                              void* d_out, int out_size, void* d_ws, size_t ws_size,
                              hipStream_t stream) {
  (void)in_sizes; (void)n_in; (void)out_size; (void)ws_size;
  const float* img1   = (const float*)d_in[0];
  const float* img2   = (const float*)d_in[1];
  const float* window = (const float*)d_in[2];
  float* out      = (float*)d_out;
  float* partials = (float*)d_ws;   // 6144 floats

  // 48 planes * 32 row-tiles * 16 col-tiles = 24576 tiles; 4 waves/block.
  const int nblocks = 6144;
  ssim_tiles_kernel<<<nblocks, 128, 0, stream>>>(img1, img2, window, partials);
  ssim_reduce_kernel<<<1, 256, 0, stream>>>(partials, nblocks, out);
}